// CapLayer_31628139168599
// MI455X (gfx1250) — compile-verified
//
#include <hip/hip_runtime.h>

typedef __attribute__((ext_vector_type(2))) float v2f;
typedef __attribute__((ext_vector_type(8))) float v8f;

#define J_NUM   10
#define D_NUM   16
#define HW      36
#define I_TOT   1152
#define EPSF    2.220446049250313e-16f

/* LDS layout (float offsets) */
#define OFF_XS    0        /* padded x tile [g][ii][48] : 32*8*48 = 12288 */
#define OFF_WS    12288    /* [g][d][ii] : 4096 */
#define OFF_BIAS  16384    /* [g][d] : 512 */
#define OFF_PRED  16896    /* [d][i] stride 1152 : 18432 */
#define OFF_BLOG  35328    /* 1152 */
#define OFF_C     36480    /* 1152 */
#define OFF_RED   37632    /* 128 */
#define OFF_S     37760    /* 16 */
#define OFF_V     37776    /* 16 */
#define OFF_COEFF 37792    /* 1 */
#define LDS_FLOATS 37808   /* ~151 KB -> still 2 workgroups per WGP (320 KB LDS) */

__device__ __forceinline__ unsigned lds_byte_addr(const void* p) {
    /* ISA: generic shared address low 32 bits ARE the LDS byte address */
    return (unsigned)(unsigned long long)p;
}

__global__ __launch_bounds__(128, 1)
void caps_route_kernel(const float* __restrict__ x,
                       const float* __restrict__ W,
                       const float* __restrict__ bias,
                       float* __restrict__ out)
{
    extern __shared__ __align__(16) float smem[];
    float* xs    = smem + OFF_XS;
    float* wsm   = smem + OFF_WS;
    float* biasm = smem + OFF_BIAS;
    float* predm = smem + OFF_PRED;
    float* blog  = smem + OFF_BLOG;
    float* cm    = smem + OFF_C;
    float* red   = smem + OFF_RED;
    float* sm    = smem + OFF_S;
    float* vm    = smem + OFF_V;
    float* coeffm= smem + OFF_COEFF;

    const int t = threadIdx.x;
    const int j = blockIdx.x;         /* 0..9  */
    const int b = blockIdx.y;         /* 0..255 */

    /* ---- Phase 1: async global->LDS staging (ASYNCcnt path) ---- */
    const unsigned long long xb = (unsigned long long)(const void*)(x + (size_t)b * 9216);
    const unsigned long long wb = (unsigned long long)(const void*)W;
    const unsigned long long sb = (unsigned long long)(const void*)bias;

    /* x[b]: 256 rows of 36 floats (144 B) -> padded LDS rows of 48 floats (192 B).
       INST_OFFSET is added to both global and LDS address, so 9 B128s share one setup. */
#pragma unroll
    for (int r = 0; r < 2; ++r) {
        const int row = t + (r << 7);
        const unsigned ldsa = lds_byte_addr(xs + row * 48);
        const unsigned voff = (unsigned)row * 144u;
        asm volatile(
            "global_load_async_to_lds_b128 %0, %1, %2 offset:0\n\t"
            "global_load_async_to_lds_b128 %0, %1, %2 offset:16\n\t"
            "global_load_async_to_lds_b128 %0, %1, %2 offset:32\n\t"
            "global_load_async_to_lds_b128 %0, %1, %2 offset:48\n\t"
            "global_load_async_to_lds_b128 %0, %1, %2 offset:64\n\t"
            "global_load_async_to_lds_b128 %0, %1, %2 offset:80\n\t"
            "global_load_async_to_lds_b128 %0, %1, %2 offset:96\n\t"
            "global_load_async_to_lds_b128 %0, %1, %2 offset:112\n\t"
            "global_load_async_to_lds_b128 %0, %1, %2 offset:128"
            :: "v"(ldsa), "v"(voff), "s"(xb) : "memory");
    }
    /* W j-slice: per g, 128 contiguous floats at g*1280 + j*128 -> dense LDS */
    for (int idx = t; idx < 1024; idx += 128) {
        const int g = idx >> 5, w4 = idx & 31;
        const unsigned ldsa = lds_byte_addr(wsm + g * 128 + w4 * 4);
        const unsigned voff = (unsigned)(g * 1280 + j * 128 + w4 * 4) * 4u;
        asm volatile("global_load_async_to_lds_b128 %0, %1, %2"
                     :: "v"(ldsa), "v"(voff), "s"(wb) : "memory");
    }
    /* bias j-slice: 512 floats, 128 B128s */
    {
        const int g = t >> 2, q = t & 3;
        const unsigned ldsa = lds_byte_addr(biasm + g * 16 + q * 4);
        const unsigned voff = (unsigned)(g * 160 + j * 16 + q * 4) * 4u;
        asm volatile("global_load_async_to_lds_b128 %0, %1, %2"
                     :: "v"(ldsa), "v"(voff), "s"(sb) : "memory");
    }
    /* zero-fill padding cols 36..47 (disjoint from async writes) + routing logits */
    for (int idx = t; idx < 3072; idx += 128) {
        const int row = idx / 12, c = idx - row * 12;
        xs[row * 48 + 36 + c] = 0.0f;
    }
    for (int i = t; i < I_TOT; i += 128) blog[i] = 0.0f;

    asm volatile("s_wait_asynccnt 0" ::: "memory");
    __syncthreads();

    /* ---- Phase 2: pred[d][i] via V_WMMA_F32_16X16X4_F32 (branch-free operands) ---- */
    const int lane = t & 31;
    const int wave = t >> 5;
    const int m    = lane & 15;       /* A row M / B col N / D col N */
    const int hi   = lane >> 4;       /* K-half for A/B, M-half for D */

    for (int tile = wave; tile < 96; tile += 4) {
        const int g  = tile / 3;
        const int st = tile % 3;
        const int scol = st * 16 + m;
        v8f acc = {};
#pragma unroll
        for (int ks = 0; ks < 2; ++ks) {
            const int kb = ks * 4 + hi * 2;
            v2f a, bv;
            a.x  = wsm[g * 128 + m * 8 + kb];
            a.y  = wsm[g * 128 + m * 8 + kb + 1];
            bv.x = xs[g * 384 + kb * 48 + scol];        /* padded: always valid */
            bv.y = xs[g * 384 + (kb + 1) * 48 + scol];
            acc = __builtin_amdgcn_wmma_f32_16x16x4_f32(
                      false, a, false, bv, (short)0, acc, false, false);
        }
        if (scol < HW) {
            const int ig = g * 36 + scol;
#pragma unroll
            for (int r = 0; r < 8; ++r) {
                const int d = r + hi * 8;
                predm[d * I_TOT + ig] = acc[r] + biasm[g * 16 + d];
            }
        }
    }
    __syncthreads();

    /* ---- Phase 3: dynamic routing, 3 iterations ---- */
    for (int it = 0; it < 3; ++it) {
        /* softmax over i: c = softmax(blog). blog==0 at it=0 -> uniform 1/I. */
        float lm = -3.402823e38f;
        for (int i = t; i < I_TOT; i += 128) lm = fmaxf(lm, blog[i]);
        red[t] = lm; __syncthreads();
        for (int off = 64; off > 0; off >>= 1) {
            if (t < off) red[t] = fmaxf(red[t], red[t + off]);
            __syncthreads();
        }
        const float bmax = red[0]; __syncthreads();

        float ls = 0.0f;
        for (int i = t; i < I_TOT; i += 128) {
            float e = __expf(blog[i] - bmax);
            cm[i] = e; ls += e;
        }
        red[t] = ls; __syncthreads();
        for (int off = 64; off > 0; off >>= 1) {
            if (t < off) red[t] += red[t + off];
            __syncthreads();
        }
        const float inv = 1.0f / red[0]; __syncthreads();
        for (int i = t; i < I_TOT; i += 128) cm[i] *= inv;
        __syncthreads();

        /* s[d] = sum_i pred[d][i] * c[i] : 8 lanes per d */
        {
            const int d = t >> 3, sub = t & 7;
            const float* pr = predm + d * I_TOT;
            float acc = 0.0f;
            for (int i = sub; i < I_TOT; i += 8) acc += pr[i] * cm[i];
            red[t] = acc;
        }
        __syncthreads();
        if (t < 16) {
            float s = 0.0f;
#pragma unroll
            for (int k = 0; k < 8; ++k) s += red[t * 8 + k];
            sm[t] = s;
        }
        __syncthreads();

        /* squash ('paper' manner) */
        if (t == 0) {
            float n2 = 0.0f;
#pragma unroll
            for (int d = 0; d < 16; ++d) n2 += sm[d] * sm[d];
            const float nrm = sqrtf(n2);
            coeffm[0] = (n2 / (1.0f + n2)) / (nrm + EPSF);
        }
        __syncthreads();
        if (t < 16) vm[t] = sm[t] * coeffm[0];
        __syncthreads();

        /* logit update: b[i] += sum_d v[d] * pred[d][i] */
        if (it < 2) {
            for (int i = t; i < I_TOT; i += 128) {
                float a = 0.0f;
#pragma unroll
                for (int d = 0; d < 16; ++d) a += vm[d] * predm[d * I_TOT + i];
                blog[i] += a;
            }
            __syncthreads();
        }
    }

    if (t < 16) out[((size_t)b * J_NUM + j) * D_NUM + t] = vm[t];
}

extern "C" void kernel_launch(void* const* d_in, const int* in_sizes, int n_in,
                              void* d_out, int out_size, void* d_ws, size_t ws_size,
                              hipStream_t stream) {
    (void)in_sizes; (void)n_in; (void)d_ws; (void)ws_size; (void)out_size;
    const float* x    = (const float*)d_in[0];
    const float* W    = (const float*)d_in[1];
    const float* bias = (const float*)d_in[2];
    float* out = (float*)d_out;

    dim3 grid(J_NUM, 256);
    dim3 block(128);
    const size_t shmem = (size_t)LDS_FLOATS * sizeof(float);
    caps_route_kernel<<<grid, block, shmem, stream>>>(x, W, bias, out);
}